// BidirectionalAttention_42623255446187
// MI455X (gfx1250) — compile-verified
//
#include <hip/hip_runtime.h>
#include <math.h>

#if defined(__HIP_DEVICE_COMPILE__) && !__has_builtin(__builtin_amdgcn_wmma_f32_16x16x4_f32)
#error "missing __builtin_amdgcn_wmma_f32_16x16x4_f32 on device toolchain"
#endif

typedef __attribute__((ext_vector_type(2))) float v2f;
typedef __attribute__((ext_vector_type(8))) float v8f;

#define BATCH 4
#define CDIM  256
#define HWN   4096      // 64*64
#define CR    32
#define OC    128
#define OH    128

// ---------------------------------------------------------------------------
// Generic row-major fp32 WMMA GEMM:  D[m][n] = sum_k A[m][k]*B[k][n]
//   A: M x K row-major (per-batch stride sA), B: K x N row-major (stride sB)
//   epilogue: *(gamma[0]*spatial[b]) if gamma, +bias[m] if bias, +res if res
// block = 256 thr (8 waves) arranged 4(M) x 2(N): block tile 256M x 32N.
// wave tile = 64(M) x 16(N). grid = (N/32, M/256, batch).
// M%256==0, N%32==0, K%4==0 assumed. B is streamed only M/256 times.
// ---------------------------------------------------------------------------
__global__ __launch_bounds__(256)
void gemm_rm_wmma(const float* __restrict__ A, long sA,
                  const float* __restrict__ B, long sB,
                  float* __restrict__ D, long sD,
                  int M, int N, int K,
                  const float* __restrict__ bias,
                  const float* __restrict__ gamma,
                  const float* __restrict__ spatial,
                  const float* __restrict__ res, long sRes)
{
#if defined(__HIP_DEVICE_COMPILE__)
    const int b = blockIdx.z;
    A += (long)b * sA;
    B += (long)b * sB;
    D += (long)b * sD;

    const int lane = threadIdx.x & 31;
    const int wave = threadIdx.x >> 5;
    const int l15  = lane & 15;
    const int h    = lane >> 4;               // half-wave: K pair select
    const int m0   = blockIdx.y * 256 + (wave >> 1) * 64;
    const int n0   = blockIdx.x * 32  + (wave & 1) * 16;

    v8f acc[4];
#pragma unroll
    for (int t = 0; t < 4; ++t) acc[t] = 0.0f;

    for (int k0 = 0; k0 < K; k0 += 4) {
        const int ka = k0 + 2 * h;            // this half-wave's K columns
        v2f bv;
        bv.x = B[(long)ka * N + n0 + l15];
        bv.y = B[(long)(ka + 1) * N + n0 + l15];
#pragma unroll
        for (int t = 0; t < 4; ++t) {
            const float* ap = A + (long)(m0 + t * 16 + l15) * K + ka;
            v2f av;
            av.x = ap[0];
            av.y = ap[1];
            acc[t] = __builtin_amdgcn_wmma_f32_16x16x4_f32(
                false, av, false, bv, (short)0, acc[t], false, false);
        }
    }

    float scale = 1.0f;
    if (gamma) scale = gamma[0] * spatial[b];

#pragma unroll
    for (int t = 0; t < 4; ++t) {
#pragma unroll
        for (int g = 0; g < 8; ++g) {
            const int  m   = m0 + t * 16 + g + 8 * h;
            const long off = (long)m * N + n0 + l15;
            float v = acc[t][g] * scale;
            if (bias) v += bias[m];
            if (res)  v += res[(long)b * sRes + off];
            D[off] = v;
        }
    }
#endif
}

// ---------------------------------------------------------------------------
// attn raw scores, stored transposed:  Sraw[b][m][n] = sum_c key[c][m]*query[c][n]
// K = 32. Same wave tiling. grid = (HWN/32, HWN/256, BATCH)
// ---------------------------------------------------------------------------
__global__ __launch_bounds__(256)
void attn_gemm_wmma(const float* __restrict__ Q, const float* __restrict__ Kv,
                    float* __restrict__ Sraw)
{
#if defined(__HIP_DEVICE_COMPILE__)
    const int b = blockIdx.z;
    Q  += (long)b * CR * HWN;
    Kv += (long)b * CR * HWN;
    float* S = Sraw + (long)b * HWN * HWN;

    const int lane = threadIdx.x & 31;
    const int wave = threadIdx.x >> 5;
    const int l15  = lane & 15;
    const int h    = lane >> 4;
    const int m0   = blockIdx.y * 256 + (wave >> 1) * 64;
    const int n0   = blockIdx.x * 32  + (wave & 1) * 16;

    v8f acc[4];
#pragma unroll
    for (int t = 0; t < 4; ++t) acc[t] = 0.0f;

#pragma unroll
    for (int k0 = 0; k0 < CR; k0 += 4) {
        const int ka = k0 + 2 * h;
        v2f bv;
        bv.x = Q[ka * HWN + n0 + l15];
        bv.y = Q[(ka + 1) * HWN + n0 + l15];
#pragma unroll
        for (int t = 0; t < 4; ++t) {
            v2f av;                           // A = key^T : A[m][k] = key[k][m]
            av.x = Kv[ka * HWN + m0 + t * 16 + l15];
            av.y = Kv[(ka + 1) * HWN + m0 + t * 16 + l15];
            acc[t] = __builtin_amdgcn_wmma_f32_16x16x4_f32(
                false, av, false, bv, (short)0, acc[t], false, false);
        }
    }

#pragma unroll
    for (int t = 0; t < 4; ++t)
#pragma unroll
        for (int g = 0; g < 8; ++g)
            S[(long)(m0 + t * 16 + g + 8 * h) * HWN + n0 + l15] = acc[t][g];
#endif
}

// ---------------------------------------------------------------------------
// stride-2 transposed conv 256->128, k=4, pad=1, via WMMA implicit GEMM.
// Per output-parity class (ph,pw): out[2i+ph, 2j+pw] needs 2x2 input taps:
//   half h=0: ih = i+ph-1, weight kh = 3-ph ; half h=1: ih = i+ph, kh = 1-ph
//   dw=0: iw = j+pw-1, kw = 3-pw            ; dw=1: iw = j+pw,   kw = 1-pw
// GEMM: M=128(oc), N=4096(i*64+j), K=256*4 (one ic per 4-wide K step).
// block = 8 waves (2M x 2N... here 2M x 2N -> 128M x 32N), grid=(128,1,b*4+par)
// Boundary taps: clamped address + select (uniform/lane masks, no divergence).
// ---------------------------------------------------------------------------
__global__ __launch_bounds__(256)
void tconv_wmma(const float* __restrict__ F, const float* __restrict__ Wc,
                const float* __restrict__ bco, float* __restrict__ out)
{
#if defined(__HIP_DEVICE_COMPILE__)
    const int bz = blockIdx.z;
    const int b  = bz >> 2;
    const int ph = (bz >> 1) & 1;
    const int pw = bz & 1;

    const int lane = threadIdx.x & 31;
    const int wave = threadIdx.x >> 5;
    const int l15  = lane & 15;
    const int h    = lane >> 4;
    const int m0   = (wave >> 1) * 64;            // M=128: waves 0..1 rows? (wave>>1)∈0..3 -> use &1
    const int mrow = (wave >> 1) & 1;             // 2 M-rows of 64 -> 128
    const int m0f  = mrow * 64;
    const int n0   = (blockIdx.x * 2 + (wave & 1)) * 16;

    (void)m0;

    const int i   = n0 >> 6;                      // output-grid row (uniform per tile)
    const int j   = (n0 & 63) + l15;              // output-grid col (per lane)
    const int ihh = i + ph - 1 + h;               // this half-wave's input row
    const bool rowok = (unsigned)ihh < 64u;
    const int iw0 = j + pw - 1;
    const int iw1 = j + pw;
    const bool c0ok = rowok & ((unsigned)iw0 < 64u);
    const bool c1ok = rowok & ((unsigned)iw1 < 64u);
    const int rbase = (rowok ? ihh : 0) * 64;
    const int f0 = rbase + (((unsigned)iw0 < 64u) ? iw0 : 0);
    const int f1 = rbase + (((unsigned)iw1 < 64u) ? iw1 : 0);
    const int khh = h ? (1 - ph) : (3 - ph);
    const int kwx = 3 - pw, kwy = 1 - pw;

    const float* Fb = F + (long)b * CDIM * HWN;

    v8f acc[4];
#pragma unroll
    for (int t = 0; t < 4; ++t) acc[t] = 0.0f;

    for (int ic = 0; ic < CDIM; ++ic) {
        const float* Fr = Fb + (long)ic * HWN;
        const float t0 = Fr[f0];
        const float t1 = Fr[f1];
        v2f bv;
        bv.x = c0ok ? t0 : 0.0f;
        bv.y = c1ok ? t1 : 0.0f;
        const float* wrow = Wc + ((long)ic * OC) * 16 + khh * 4;
#pragma unroll
        for (int t = 0; t < 4; ++t) {
            const float* wp = wrow + (long)(m0f + t * 16 + l15) * 16;
            v2f av;
            av.x = wp[kwx];
            av.y = wp[kwy];
            acc[t] = __builtin_amdgcn_wmma_f32_16x16x4_f32(
                false, av, false, bv, (short)0, acc[t], false, false);
        }
    }

    const int oh = 2 * i + ph;
    const int ow = 2 * j + pw;
#pragma unroll
    for (int t = 0; t < 4; ++t) {
#pragma unroll
        for (int g = 0; g < 8; ++g) {
            const int m = m0f + t * 16 + g + 8 * h;
            out[(((long)b * OC + m) * OH + oh) * OH + ow] = acc[t][g] + bco[m];
        }
    }
#endif
}

// ---------------------------------------------------------------------------
// spatial[b] = mean(x[b])
// ---------------------------------------------------------------------------
__global__ __launch_bounds__(256)
void spatial_mean(const float* __restrict__ x, float* __restrict__ spatial)
{
    const int b = blockIdx.x;
    const float* xb = x + (long)b * CDIM * HWN;
    double s = 0.0;
    for (int i = threadIdx.x; i < CDIM * HWN; i += 256) s += (double)xb[i];
    __shared__ double red[256];
    red[threadIdx.x] = s;
    __syncthreads();
    for (int off = 128; off > 0; off >>= 1) {
        if (threadIdx.x < off) red[threadIdx.x] += red[threadIdx.x + off];
        __syncthreads();
    }
    if (threadIdx.x == 0)
        spatial[b] = (float)(red[0] / (double)(CDIM * HWN));
}

// ---------------------------------------------------------------------------
// grouped conv1d: t[b,g,p] = b1[g] + sum_{i<8,d<3} w1[g,i,d]*qlin[b,8g+i,p+d-1]
// ---------------------------------------------------------------------------
__global__ __launch_bounds__(256)
void conv_adj1(const float* __restrict__ qlin, const float* __restrict__ w1,
               const float* __restrict__ b1, float* __restrict__ t)
{
    const long idx = (long)blockIdx.x * 256 + threadIdx.x;   // B*32*HWN
    const int p = (int)(idx & (HWN - 1));
    const int g = (int)((idx >> 12) & 31);
    const int b = (int)(idx >> 17);
    const float* src = qlin + ((long)b * CDIM + g * 8) * HWN;
    const float* w = w1 + g * 24;
    float acc = b1[g];
#pragma unroll
    for (int i = 0; i < 8; ++i) {
        const float* row = src + (long)i * HWN;
        const float xm = (p > 0)       ? row[p - 1] : 0.0f;
        const float x0 = row[p];
        const float xp = (p < HWN - 1) ? row[p + 1] : 0.0f;
        acc += w[i * 3 + 0] * xm + w[i * 3 + 1] * x0 + w[i * 3 + 2] * xp;
    }
    t[idx] = acc;
}

// ---------------------------------------------------------------------------
// conv1d 32->64 + deinterleave into query (even o) / key (odd o)
// ---------------------------------------------------------------------------
__global__ __launch_bounds__(256)
void conv_adj2(const float* __restrict__ t, const float* __restrict__ w2,
               const float* __restrict__ b2, float* __restrict__ query,
               float* __restrict__ key)
{
    const long idx = (long)blockIdx.x * 256 + threadIdx.x;   // B*64*HWN
    const int p = (int)(idx & (HWN - 1));
    const int o = (int)((idx >> 12) & 63);
    const int b = (int)(idx >> 18);
    const float* src = t + (long)b * CR * HWN;
    const float* w = w2 + o * 96;
    float acc = b2[o];
    for (int g = 0; g < 32; ++g) {
        const float* row = src + (long)g * HWN;
        const float xm = (p > 0)       ? row[p - 1] : 0.0f;
        const float x0 = row[p];
        const float xp = (p < HWN - 1) ? row[p + 1] : 0.0f;
        acc += w[g * 3 + 0] * xm + w[g * 3 + 1] * x0 + w[g * 3 + 2] * xp;
    }
    float* dst = (o & 1) ? key : query;
    dst[((long)b * CR + (o >> 1)) * HWN + p] = acc;
}

// ---------------------------------------------------------------------------
// per-row (over n) softmax stats of Sraw[b][m][:]  (softmax axis=1)
// one block per row; stats[row] = {max, sumexp}
// ---------------------------------------------------------------------------
__global__ __launch_bounds__(256)
void row_softmax_stats(const float* __restrict__ S, float* __restrict__ stats)
{
    const long row = blockIdx.x;            // b*HWN + m
    const float* r = S + row * (long)HWN;
    __shared__ float red[256];
    float mx = -INFINITY;
    for (int i = threadIdx.x; i < HWN; i += 256) mx = fmaxf(mx, r[i]);
    red[threadIdx.x] = mx;
    __syncthreads();
    for (int off = 128; off > 0; off >>= 1) {
        if (threadIdx.x < off)
            red[threadIdx.x] = fmaxf(red[threadIdx.x], red[threadIdx.x + off]);
        __syncthreads();
    }
    mx = red[0];
    __syncthreads();
    float s = 0.0f;
    for (int i = threadIdx.x; i < HWN; i += 256) s += __expf(r[i] - mx);
    red[threadIdx.x] = s;
    __syncthreads();
    for (int off = 128; off > 0; off >>= 1) {
        if (threadIdx.x < off) red[threadIdx.x] += red[threadIdx.x + off];
        __syncthreads();
    }
    if (threadIdx.x == 0) {
        stats[row * 2]     = mx;
        stats[row * 2 + 1] = red[0];
    }
}

// ---------------------------------------------------------------------------
// S[b][m][n] <- softmax_over_b(Sraw)[b][m][n] + softmax_over_n(Sraw)[b][m][n]
// one thread handles one (m,n) for all 4 batches (in-place)
// ---------------------------------------------------------------------------
__global__ __launch_bounds__(256)
void combine_softmax(float* __restrict__ S, const float* __restrict__ stats)
{
    const long idx = (long)blockIdx.x * 256 + threadIdx.x;   // over HWN*HWN
    const long m  = idx >> 12;
    const long plane = (long)HWN * HWN;
    float v[BATCH];
#pragma unroll
    for (int b = 0; b < BATCH; ++b) v[b] = S[idx + (long)b * plane];
    float mb = fmaxf(fmaxf(v[0], v[1]), fmaxf(v[2], v[3]));
    float e[BATCH];
    float es = 0.0f;
#pragma unroll
    for (int b = 0; b < BATCH; ++b) { e[b] = __expf(v[b] - mb); es += e[b]; }
    const float inv = 1.0f / es;
#pragma unroll
    for (int b = 0; b < BATCH; ++b) {
        const long row = (long)b * HWN + m;
        const float ab = __expf(v[b] - stats[row * 2]) / stats[row * 2 + 1];
        S[idx + (long)b * plane] = e[b] * inv + ab;
    }
}

// ---------------------------------------------------------------------------
extern "C" void kernel_launch(void* const* d_in, const int* in_sizes, int n_in,
                              void* d_out, int out_size, void* d_ws, size_t ws_size,
                              hipStream_t stream)
{
    const float* x     = (const float*)d_in[0];
    const float* wq    = (const float*)d_in[1];
    const float* bq    = (const float*)d_in[2];
    const float* wv    = (const float*)d_in[3];
    const float* bv    = (const float*)d_in[4];
    const float* w1    = (const float*)d_in[5];
    const float* b1    = (const float*)d_in[6];
    const float* w2    = (const float*)d_in[7];
    const float* b2    = (const float*)d_in[8];
    const float* gamma = (const float*)d_in[9];
    const float* w_co  = (const float*)d_in[10];
    const float* b_co  = (const float*)d_in[11];

    float* ws     = (float*)d_ws;
    float* qlin   = ws;                                   // B*C*HWN
    float* vbuf   = qlin   + (long)BATCH * CDIM * HWN;    // B*C*HWN
    float* tbuf   = vbuf   + (long)BATCH * CDIM * HWN;    // B*CR*HWN
    float* query  = tbuf   + (long)BATCH * CR * HWN;      // B*CR*HWN
    float* key    = query  + (long)BATCH * CR * HWN;      // B*CR*HWN
    float* fus    = key    + (long)BATCH * CR * HWN;      // B*C*HWN
    float* spat   = fus    + (long)BATCH * CDIM * HWN;    // 64 (padded)
    float* stats  = spat   + 64;                          // B*HWN*2
    float* S      = stats  + (long)BATCH * HWN * 2;       // B*HWN*HWN

    const long sCHW = (long)CDIM * HWN;

    // 0) spatial means
    spatial_mean<<<BATCH, 256, 0, stream>>>(x, spat);

    // 1) q = wq @ x + bq ; v = wv @ x + bv   (256M x 32N blocks)
    dim3 gProj(HWN / 32, CDIM / 256, BATCH);
    gemm_rm_wmma<<<gProj, 256, 0, stream>>>(wq, 0, x, sCHW, qlin, sCHW,
                                            CDIM, HWN, CDIM, bq,
                                            nullptr, nullptr, nullptr, 0);
    gemm_rm_wmma<<<gProj, 256, 0, stream>>>(wv, 0, x, sCHW, vbuf, sCHW,
                                            CDIM, HWN, CDIM, bv,
                                            nullptr, nullptr, nullptr, 0);

    // 2) adjustment convs -> query / key  (each B*CR*HWN)
    conv_adj1<<<(BATCH * CR * HWN) / 256, 256, 0, stream>>>(qlin, w1, b1, tbuf);
    conv_adj2<<<(BATCH * 2 * CR * HWN) / 256, 256, 0, stream>>>(tbuf, w2, b2, query, key);

    // 3) raw attention scores, stored transposed: S[b][m][n] = attn[b,n,m]
    dim3 gAttn(HWN / 32, HWN / 256, BATCH);
    attn_gemm_wmma<<<gAttn, 256, 0, stream>>>(query, key, S);

    // 4) softmax(axis=1 of attn) row stats + combine with softmax(axis=0)
    row_softmax_stats<<<BATCH * HWN, 256, 0, stream>>>(S, stats);
    combine_softmax<<<(int)(((long)HWN * HWN) / 256), 256, 0, stream>>>(S, stats);

    // 5) fusion = value @ S, scaled by gamma*spatial[b], + x residual
    //    (S streamed from HBM exactly once: grid.y == 1)
    dim3 gFus(HWN / 32, CDIM / 256, BATCH);
    gemm_rm_wmma<<<gFus, 256, 0, stream>>>(vbuf, sCHW, S, (long)HWN * HWN,
                                           fus, sCHW, CDIM, HWN, HWN, nullptr,
                                           gamma, spat, x, sCHW);

    // 6) transposed conv upsample via WMMA -> out (B,128,128,128)
    dim3 gT(HWN / 32, 1, BATCH * 4);
    tconv_wmma<<<gT, 256, 0, stream>>>(fus, w_co, b_co, (float*)d_out);

    (void)in_sizes; (void)n_in; (void)out_size; (void)ws_size;
}